// CRFuse_89988154786457
// MI455X (gfx1250) — compile-verified
//
#include <hip/hip_runtime.h>
#include <hip/hip_bf16.h>

// CRF + projection for MI455X (gfx1250, wave32).
// Inputs (setup_inputs order): x(B,T,H) f32, label(B,T) i32, seqlen(B) i32,
// W(H,L) f32, b(L) f32, cls_embed(2,H) f32, transitions(L,L) f32.
// Output: viterbi (B*T) as float, then loss (1 float).

#define Bn 64
#define Tn 512
#define Hn 768
#define Ln 9
#define NP 16               // L padded to 16 for WMMA N dimension
#define BT (Bn * Tn)        // 32768 rows of the GEMM
#define N2 (2 * Bn)         // 128 CRF sequences

typedef __attribute__((ext_vector_type(2))) float v2f;
typedef __attribute__((ext_vector_type(8))) float v8f;

// ---- workspace layout (floats) ----
// [0,                524288)  yws   : x@W, padded to 16 cols (2 MB)
// [524288,           524320)  cbias : 2 x 16 (cls@W + b, zero padded)
// [524320,           524448)  ll    : per-sequence log-likelihood (128)
// [524448, +65536 ints)       vit   : Viterbi paths for 2B sequences
#define WS_CBIAS 524288
#define WS_LL    524320
#define WS_VIT   524448

// -------- tiny prep: cbias[h][l] = cls[h] @ W[:,l] + b[l] --------
__global__ void crf_prep_cbias(const float* __restrict__ W,
                               const float* __restrict__ bvec,
                               const float* __restrict__ cls,
                               float* __restrict__ cbias) {
    int tid = threadIdx.x;            // 32 threads: (h = tid>>4, l = tid&15)
    int h = tid >> 4, l = tid & 15;
    float acc = 0.f;
    if (l < Ln) {
        const float* c = cls + h * Hn;
        for (int k = 0; k < Hn; ++k) acc += c[k] * W[k * Ln + l];
        acc += bvec[l];
    }
    cbias[tid] = acc;
}

// -------- memory-bound GEMM via fp32 WMMA: yws = x @ Wpad --------
// 4 waves/block, each wave computes one 16x16 tile (16 rows, K=768 in steps of 4).
__global__ void __launch_bounds__(128)
crf_gemm_pot(const float* __restrict__ x, const float* __restrict__ W,
             float* __restrict__ yws) {
    __shared__ float Wl[Hn * NP];     // 48 KB: W zero-padded to 16 columns
    int tid = threadIdx.x;
    for (int i = tid; i < Hn * NP; i += 128) {
        int r = i >> 4, c = i & 15;
        Wl[i] = (c < Ln) ? W[r * Ln + c] : 0.f;
    }
    __syncthreads();

    int wave = tid >> 5, lane = tid & 31;
    int rowBase = (blockIdx.x * 4 + wave) * 16;
    int m  = lane & 15;               // row (A) / col (B,D) within tile
    int hi = lane >> 4;               // 0: K=k..k+1, 1: K=k+2..k+3

    const float* arow = x + (size_t)(rowBase + m) * Hn;
    v8f acc = {0.f, 0.f, 0.f, 0.f, 0.f, 0.f, 0.f, 0.f};
    for (int k = 0; k < Hn; k += 4) {
        v2f a, bfr;
        a.x   = arow[k + 2 * hi];                 // A[m][k+2*hi]
        a.y   = arow[k + 2 * hi + 1];             // A[m][k+2*hi+1]
        bfr.x = Wl[(k + 2 * hi)     * NP + m];    // B[k+2*hi][m]
        bfr.y = Wl[(k + 2 * hi + 1) * NP + m];    // B[k+2*hi+1][m]
        acc = __builtin_amdgcn_wmma_f32_16x16x4_f32(
            /*neg_a=*/false, a, /*neg_b=*/false, bfr,
            /*c_mod=*/(short)0, acc, /*reuse_a=*/false, /*reuse_b=*/false);
    }
    // D: VGPR r -> (row rowBase+r, col m) for lanes 0-15; row+8 for lanes 16-31
    float* yo = yws + (size_t)(rowBase + 8 * hi) * NP + m;
    #pragma unroll
    for (int r = 0; r < 8; ++r) yo[(size_t)r * NP] = acc[r];
}

// -------- CRF forward + Viterbi: one wave32 per sequence --------
__global__ void __launch_bounds__(32)
crf_scan(const float* __restrict__ yws, const float* __restrict__ cbias,
         const int* __restrict__ label, const int* __restrict__ seqlen,
         const float* __restrict__ trans, int* __restrict__ vit_ws,
         float* __restrict__ ll_ws) {
    __shared__ unsigned char bp[Tn * Ln];   // backpointers, 4608 B

    int n    = blockIdx.x;            // 0..127
    int half = (n >= Bn) ? 1 : 0;
    int b    = n & (Bn - 1);
    int lane = threadIdx.x;
    int j    = (lane < Ln) ? lane : (Ln - 1);   // state owned by this lane

    float trc[Ln];                    // transition column j: Tr[i][j]
    #pragma unroll
    for (int i = 0; i < Ln; ++i) trc[i] = trans[i * Ln + j];
    float cb = cbias[half * NP + j];
    int   sl = seqlen[b];
    const float* Y = yws + (size_t)b * Tn * NP;

    float alpha  = Y[j] + cb;         // pot[n][0][j]
    float vstate = alpha;

    for (int t = 1; t < Tn; ++t) {
        float p = Y[t * NP + j] + cb;
        bool active = (t < sl);       // uniform across the wave
        float av[Ln], vv[Ln];
        #pragma unroll
        for (int i = 0; i < Ln; ++i) {
            av[i] = __shfl(alpha,  i, 32) + trc[i];
            vv[i] = __shfl(vstate, i, 32) + trc[i];
        }
        float mx = av[0];
        #pragma unroll
        for (int i = 1; i < Ln; ++i) mx = fmaxf(mx, av[i]);
        float s = 0.f;
        #pragma unroll
        for (int i = 0; i < Ln; ++i) s += __expf(av[i] - mx);
        float bestv = vv[0]; int bi = 0;
        #pragma unroll
        for (int i = 1; i < Ln; ++i) if (vv[i] > bestv) { bestv = vv[i]; bi = i; }
        if (active) { alpha = mx + __logf(s) + p; vstate = bestv + p; }
        else        { bi = j; }       // identity backpointer when masked
        if (lane < Ln) bp[t * Ln + j] = (unsigned char)bi;
    }

    // lognorm = logsumexp(alpha), last = argmax(vstate) (first max wins)
    float av2[Ln], vv2[Ln];
    #pragma unroll
    for (int i = 0; i < Ln; ++i) {
        av2[i] = __shfl(alpha,  i, 32);
        vv2[i] = __shfl(vstate, i, 32);
    }
    float mx = av2[0];
    #pragma unroll
    for (int i = 1; i < Ln; ++i) mx = fmaxf(mx, av2[i]);
    float s = 0.f;
    #pragma unroll
    for (int i = 0; i < Ln; ++i) s += __expf(av2[i] - mx);
    float lognorm = mx + __logf(s);
    float bm = vv2[0]; int last = 0;
    #pragma unroll
    for (int i = 1; i < Ln; ++i) if (vv2[i] > bm) { bm = vv2[i]; last = i; }

    // unary + transition gold scores (lane-parallel over t)
    const int* lab = label + b * Tn;
    float usum = 0.f, tsum = 0.f;
    for (int t = lane; t < Tn; t += 32) {
        if (t < sl) {
            int raw = lab[t];
            int lr  = half ? (raw > 0 ? raw + 1 : 0) : (raw > 0 ? 1 : 0);
            usum += Y[t * NP + lr] + cbias[half * NP + lr];
            if (t >= 1) {
                int rawp = lab[t - 1];
                int lp   = half ? (rawp > 0 ? rawp + 1 : 0) : (rawp > 0 ? 1 : 0);
                tsum += trans[lp * Ln + lr];
            }
        }
    }
    #pragma unroll
    for (int off = 16; off >= 1; off >>= 1) {
        usum += __shfl_xor(usum, off, 32);
        tsum += __shfl_xor(tsum, off, 32);
    }
    if (lane == 0) ll_ws[n] = usum + tsum - lognorm;

    __syncthreads();                  // make bp[] visible before backtrack
    if (lane == 0) {
        int tag = last;
        int* vo = vit_ws + n * Tn;
        vo[Tn - 1] = ((Tn - 1) < sl) ? tag : 0;
        for (int t = Tn - 1; t >= 1; --t) {
            tag = bp[t * Ln + tag];
            vo[t - 1] = ((t - 1) < sl) ? tag : 0;
        }
    }
}

// -------- combine the two halves' Viterbi paths --------
__global__ void crf_finalize_vit(const int* __restrict__ vit_ws,
                                 float* __restrict__ out) {
    int idx = blockIdx.x * blockDim.x + threadIdx.x;   // over B*T
    if (idx >= BT) return;
    int b = idx / Tn, t = idx % Tn;
    int v1 = vit_ws[b * Tn + t];
    int v2 = vit_ws[(Bn + b) * Tn + t];
    int o  = (v1 == 0 || v2 == 0) ? 0 : (v2 - 1);
    out[idx] = (float)o;
}

// -------- loss = mean(-ll) over 2B sequences --------
__global__ void crf_finalize_loss(const float* __restrict__ ll_ws,
                                  float* __restrict__ out) {
    int lane = threadIdx.x;           // 32 threads
    float sacc = 0.f;
    for (int i = lane; i < N2; i += 32) sacc -= ll_ws[i];
    #pragma unroll
    for (int off = 16; off >= 1; off >>= 1) sacc += __shfl_xor(sacc, off, 32);
    if (lane == 0) out[BT] = sacc / (float)N2;
}

extern "C" void kernel_launch(void* const* d_in, const int* in_sizes, int n_in,
                              void* d_out, int out_size, void* d_ws, size_t ws_size,
                              hipStream_t stream) {
    (void)in_sizes; (void)n_in; (void)out_size; (void)ws_size;
    const float* x      = (const float*)d_in[0];
    const int*   label  = (const int*)  d_in[1];
    const int*   seqlen = (const int*)  d_in[2];
    const float* W      = (const float*)d_in[3];
    const float* bvec   = (const float*)d_in[4];
    const float* cls    = (const float*)d_in[5];
    const float* trans  = (const float*)d_in[6];
    float* out = (float*)d_out;

    float* ws    = (float*)d_ws;
    float* yws   = ws;
    float* cbias = ws + WS_CBIAS;
    float* ll    = ws + WS_LL;
    int*   vit   = (int*)(ws + WS_VIT);

    crf_prep_cbias<<<1, 32, 0, stream>>>(W, bvec, cls, cbias);
    crf_gemm_pot<<<BT / 64, 128, 0, stream>>>(x, W, yws);   // 512 blocks x 4 waves x 16 rows
    crf_scan<<<N2, 32, 0, stream>>>(yws, cbias, label, seqlen, trans, vit, ll);
    crf_finalize_vit<<<BT / 256, 256, 0, stream>>>(vit, out);
    crf_finalize_loss<<<1, 32, 0, stream>>>(ll, out);
}